// AAMSoftmax_40699110097394
// MI455X (gfx1250) — compile-verified
//
#include <hip/hip_runtime.h>
#include <hip/hip_bf16.h>
#include <math.h>

// ---------------------------------------------------------------------------
// AAM-Softmax (ArcFace) forward, fused GEMM + online softmax for MI455X.
//
// GEMM: [1024 x 256] x [256 x 100000] f32 via V_WMMA_F32_16X16X4_F32.
// 52.4 GFLOP; weight = 102.4 MB (L2-resident, 192 MB). Logits (400 MB) never
// materialized: margin + online softmax + argmax + target logit fused per
// (row, class-chunk); partials reduced afterwards.
// Weight tiles are double-buffered in LDS via async global->LDS copies
// (GLOBAL_LOAD_ASYNC_TO_LDS_B128 / ASYNCcnt) when the toolchain exposes them.
// ---------------------------------------------------------------------------

typedef __attribute__((ext_vector_type(2))) float v2f;
typedef __attribute__((ext_vector_type(8))) float v8f;
typedef int v4i __attribute__((vector_size(16)));   // matches builtin signature

#define EMBED_DIM   256
#define NUM_CLASSES 100000
#define BATCHSZ     1024
#define NCHUNK      125          // class chunks
#define CPC         800          // classes per chunk (125*800 = 100000)
#define ROWS_PB     64           // batch rows per block (4 waves x 16)
#define NTILE       16
#define TILE_ELEMS  (NTILE * EMBED_DIM)   // 4096 floats = 16 KB

#define AAM_SCALE   30.0f
#define COS_M       0.9800665778412416f   // cos(0.2)
#define SIN_M       0.19866933079506122f  // sin(0.2)
#define CLIP_EPS    1e-7f
#define NEG_BIG     (-1.0e30f)

#if __has_builtin(__builtin_amdgcn_global_load_async_to_lds_b128) && \
    __has_builtin(__builtin_amdgcn_s_wait_asynccnt)
#define HAS_ASYNC_LDS 1
#else
#define HAS_ASYNC_LDS 0
#endif

#if HAS_ASYNC_LDS
typedef __attribute__((address_space(1))) v4i g_v4i;
typedef __attribute__((address_space(3))) v4i l_v4i;

template <int OFF>
__device__ __forceinline__ void async_b128(const float* g, float* l) {
    // INST_OFFSET is added to BOTH the global and LDS addresses (ISA 8 §4.4)
    __builtin_amdgcn_global_load_async_to_lds_b128(
        (g_v4i*)g, (l_v4i*)l, OFF, 0);
}

// Copy one 16x256 f32 weight tile (16 KB) to LDS: 128 threads x 8 x b128.
// float4 index i = tid + 128*j  ->  byte offset tid*16 + j*2048 in both spaces
// (weight rows for consecutive classes are contiguous in memory).
__device__ __forceinline__ void issue_tile(float* sDst, const float* wSrc, int tid) {
    const float* g = wSrc + tid * 4;
    float*       l = sDst + tid * 4;
    async_b128<0>(g, l);
    async_b128<2048>(g, l);
    async_b128<4096>(g, l);
    async_b128<6144>(g, l);
    async_b128<8192>(g, l);
    async_b128<10240>(g, l);
    async_b128<12288>(g, l);
    async_b128<14336>(g, l);
}
#endif

// ---------------- kernel 1: normalize embedding rows -----------------------
__global__ __launch_bounds__(256)
void aam_norm_emb(const float* __restrict__ X, float* __restrict__ Y) {
    int wave = threadIdx.x >> 5;
    int lane = threadIdx.x & 31;
    int row  = blockIdx.x * 8 + wave;                 // 128 blocks * 8 waves
    const float4* x4 = (const float4*)(X + (size_t)row * EMBED_DIM);
    float4 a = x4[lane];
    float4 b = x4[lane + 32];
    float ss = a.x*a.x + a.y*a.y + a.z*a.z + a.w*a.w
             + b.x*b.x + b.y*b.y + b.z*b.z + b.w*b.w;
    #pragma unroll
    for (int d = 1; d < 32; d <<= 1) ss += __shfl_xor(ss, d, 32);
    float inv = 1.0f / fmaxf(sqrtf(ss), 1e-12f);
    float4* y4 = (float4*)(Y + (size_t)row * EMBED_DIM);
    float4 oa; oa.x=a.x*inv; oa.y=a.y*inv; oa.z=a.z*inv; oa.w=a.w*inv;
    float4 ob; ob.x=b.x*inv; ob.y=b.y*inv; ob.z=b.z*inv; ob.w=b.w*inv;
    y4[lane] = oa;
    y4[lane + 32] = ob;
}

// ---------------- kernel 2: per-class inverse norms ------------------------
__global__ __launch_bounds__(256)
void aam_norm_w(const float* __restrict__ W, float* __restrict__ winv) {
    int wave = threadIdx.x >> 5;
    int lane = threadIdx.x & 31;
    int row  = blockIdx.x * 8 + wave;                 // 12500 blocks
    const float4* x4 = (const float4*)(W + (size_t)row * EMBED_DIM);
    float4 a = x4[lane];
    float4 b = x4[lane + 32];
    float ss = a.x*a.x + a.y*a.y + a.z*a.z + a.w*a.w
             + b.x*b.x + b.y*b.y + b.z*b.z + b.w*b.w;
    #pragma unroll
    for (int d = 1; d < 32; d <<= 1) ss += __shfl_xor(ss, d, 32);
    if (lane == 0) winv[row] = 1.0f / fmaxf(sqrtf(ss), 1e-12f);
}

// ---------------- kernel 3: fused WMMA GEMM + online softmax ---------------
__global__ __launch_bounds__(128)
void aam_main(const float* __restrict__ embn,
              const float* __restrict__ weight,
              const float* __restrict__ winv,
              const int*   __restrict__ labels,
              float*       __restrict__ partials) {
    __shared__ __align__(16) float sA[ROWS_PB * EMBED_DIM];   // 64 KB
    __shared__ __align__(16) float sB[2 * TILE_ELEMS];        // 2 x 16 KB
    __shared__ int sLab[ROWS_PB];

    const int tid       = threadIdx.x;            // 0..127 (4 waves)
    const int chunk     = blockIdx.x;             // 0..124
    const int rowBase   = blockIdx.y * ROWS_PB;   // 16 batch tiles
    const int classBase = chunk * CPC;

#if HAS_ASYNC_LDS
    // kick off the first weight tile before staging A (overlap)
    issue_tile(sB, weight + (size_t)classBase * EMBED_DIM, tid);
#endif

    // stage normalized embedding tile (64 x 256 f32)
    {
        const float4* src = (const float4*)(embn + (size_t)rowBase * EMBED_DIM);
        float4* dst = (float4*)sA;
        for (int i = tid; i < ROWS_PB * EMBED_DIM / 4; i += 128) dst[i] = src[i];
    }
    if (tid < ROWS_PB) sLab[tid] = labels[rowBase + tid];
    __syncthreads();

    const int wave  = tid >> 5;
    const int lane  = tid & 31;
    const int laneN = lane & 15;
    const int hi    = lane >> 4;           // 0: K {k,k+1}, rows 0-7; 1: K {k+2,k+3}, rows 8-15
    const int mrow0 = wave * NTILE;        // wave's 16 rows within the tile

    // per-lane running state for 8 rows (m = hi*8 + j), classes n == laneN (mod 16)
    float rmax[8], rsum[8], amax[8], tlog[8];
    int   aidx[8];
    #pragma unroll
    for (int j = 0; j < 8; ++j) {
        rmax[j] = NEG_BIG; rsum[j] = 0.0f;
        amax[j] = NEG_BIG; aidx[j] = 0; tlog[j] = NEG_BIG;
    }

    const float* aLane  = sA + (size_t)(mrow0 + laneN) * EMBED_DIM + hi * 2;
    const float* bLane0 = sB + (size_t)laneN * EMBED_DIM + hi * 2;
    const float* bLane1 = bLane0 + TILE_ELEMS;

    for (int nt = 0, t = 0; nt < CPC; nt += NTILE, ++t) {
#if HAS_ASYNC_LDS
        const float* bLane = (t & 1) ? bLane1 : bLane0;
        __builtin_amdgcn_s_wait_asynccnt(0);   // this wave's tile t slice resident
        __syncthreads();                        // all slices resident; prev reads done
        if (nt + NTILE < CPC)
            issue_tile(sB + ((t + 1) & 1) * TILE_ELEMS,
                       weight + (size_t)(classBase + nt + NTILE) * EMBED_DIM, tid);
#else
        const float* bLane = bLane0;
        for (int i = tid; i < TILE_ELEMS / 4; i += 128) {
            int n = i >> 6;                               // 64 float4 per class row
            int c = classBase + nt + n;
            ((float4*)sB)[i] =
                ((const float4*)(weight + (size_t)c * EMBED_DIM))[i & 63];
        }
        __syncthreads();
#endif

        // 16x16 tile of emb_n . w_raw: 64 chained f32 WMMAs over K=256
        v8f acc = {};
        #pragma unroll
        for (int k = 0; k < EMBED_DIM; k += 4) {
            v2f a = *(const v2f*)(aLane + k);
            v2f b = *(const v2f*)(bLane + k);
            acc = __builtin_amdgcn_wmma_f32_16x16x4_f32(
                false, a, false, b, (short)0, acc, false, false);
        }

        // epilogue: class-norm scale -> clip -> margin (target) -> online stats
        const int   c  = classBase + nt + laneN;
        const float wv = winv[c];
        #pragma unroll
        for (int j = 0; j < 8; ++j) {
            const int m = hi * 8 + j;
            float x = acc[j] * wv;
            x = fminf(fmaxf(x, -1.0f + CLIP_EPS), 1.0f - CLIP_EPS);
            const bool isT = (sLab[mrow0 + m] == c);
            float logit;
            if (isT) {
                float s = sqrtf(fmaxf(0.0f, 1.0f - x * x));   // sin(theta)
                logit = AAM_SCALE * (COS_M * x - SIN_M * s);  // cos(theta + m)
                tlog[j] = logit;
            } else {
                logit = AAM_SCALE * x;
            }
            if (logit > amax[j]) { amax[j] = logit; aidx[j] = c; }
            if (logit > rmax[j]) {
                rsum[j] = rsum[j] * __expf(rmax[j] - logit) + 1.0f;
                rmax[j] = logit;
            } else {
                rsum[j] += __expf(logit - rmax[j]);
            }
        }
#if !HAS_ASYNC_LDS
        __syncthreads();   // protect sB before next tile load
#endif
    }

    // combine the 16 lanes sharing each row, emit per-(row, chunk) partials
    #pragma unroll
    for (int j = 0; j < 8; ++j) {
        float M = rmax[j], S = rsum[j], AV = amax[j], TL = tlog[j];
        int AI = aidx[j];
        #pragma unroll
        for (int d = 1; d < 16; d <<= 1) {
            float M2  = __shfl_xor(M, d, 32);
            float S2  = __shfl_xor(S, d, 32);
            float AV2 = __shfl_xor(AV, d, 32);
            float TL2 = __shfl_xor(TL, d, 32);
            int   AI2 = __shfl_xor(AI, d, 32);
            float nM = fmaxf(M, M2);
            S = S * __expf(M - nM) + S2 * __expf(M2 - nM);
            M = nM;
            if (AV2 > AV || (AV2 == AV && AI2 < AI)) { AV = AV2; AI = AI2; }
            TL = fmaxf(TL, TL2);
        }
        if (laneN == 0) {
            const int row = rowBase + mrow0 + hi * 8 + j;
            float* p = partials + ((size_t)row * NCHUNK + chunk) * 5;
            p[0] = M; p[1] = S; p[2] = AV;
            p[3] = __int_as_float(AI); p[4] = TL;
        }
    }
}

// ---------------- kernel 4: per-row reduction over chunks ------------------
__global__ __launch_bounds__(256)
void aam_rowreduce(const float* __restrict__ partials,
                   const int*   __restrict__ labels,
                   float*       __restrict__ rowout) {
    int row = blockIdx.x * 256 + threadIdx.x;
    if (row >= BATCHSZ) return;
    float M = NEG_BIG, S = 0.0f, AV = NEG_BIG, TL = NEG_BIG;
    int AI = 0;
    for (int ch = 0; ch < NCHUNK; ++ch) {
        const float* p = partials + ((size_t)row * NCHUNK + ch) * 5;
        float M2 = p[0], S2 = p[1], AV2 = p[2], TL2 = p[4];
        int AI2 = __float_as_int(p[3]);
        float nM = fmaxf(M, M2);
        S = S * expf(M - nM) + S2 * expf(M2 - nM);
        M = nM;
        if (AV2 > AV || (AV2 == AV && AI2 < AI)) { AV = AV2; AI = AI2; }
        TL = fmaxf(TL, TL2);
    }
    float logp = TL - M - logf(S);               // log_softmax at target
    rowout[row]           = -logp;               // per-row loss
    rowout[BATCHSZ + row] = (AI == labels[row]) ? 1.0f : 0.0f;
}

// ---------------- kernel 5: final means ------------------------------------
__global__ __launch_bounds__(256)
void aam_final(const float* __restrict__ rowout, float* __restrict__ out) {
    __shared__ float sl[256], sa[256];
    int t = threadIdx.x;
    float L = 0.0f, A = 0.0f;
    for (int r = t; r < BATCHSZ; r += 256) {
        L += rowout[r];
        A += rowout[BATCHSZ + r];
    }
    sl[t] = L; sa[t] = A;
    __syncthreads();
    for (int s = 128; s > 0; s >>= 1) {
        if (t < s) { sl[t] += sl[t + s]; sa[t] += sa[t + s]; }
        __syncthreads();
    }
    if (t == 0) {
        out[0] = sl[0] / (float)BATCHSZ;   // loss
        out[1] = sa[0] / (float)BATCHSZ;   // accuracy
    }
}

// ---------------------------------------------------------------------------
extern "C" void kernel_launch(void* const* d_in, const int* in_sizes, int n_in,
                              void* d_out, int out_size, void* d_ws, size_t ws_size,
                              hipStream_t stream) {
    const float* emb    = (const float*)d_in[0];   // [1024, 256]
    const float* weight = (const float*)d_in[1];   // [100000, 256]
    const int*   labels = (const int*)  d_in[2];   // [1024]
    float* out = (float*)d_out;                    // [loss, acc]

    // workspace layout (floats): ~4.02 MB total
    float* embn     = (float*)d_ws;                            // 1024*256
    float* winv     = embn + (size_t)BATCHSZ * EMBED_DIM;      // 100000
    float* partials = winv + NUM_CLASSES;                      // 1024*125*5
    float* rowout   = partials + (size_t)BATCHSZ * NCHUNK * 5; // 2*1024

    aam_norm_emb<<<BATCHSZ / 8, 256, 0, stream>>>(emb, embn);
    aam_norm_w<<<NUM_CLASSES / 8, 256, 0, stream>>>(weight, winv);
    aam_main<<<dim3(NCHUNK, BATCHSZ / ROWS_PB), 128, 0, stream>>>(
        embn, weight, winv, labels, partials);
    aam_rowreduce<<<(BATCHSZ + 255) / 256, 256, 0, stream>>>(partials, labels, rowout);
    aam_final<<<1, 256, 0, stream>>>(rowout, out);
}